// EMGNN_51573967290496
// MI455X (gfx1250) — compile-verified
//
#include <hip/hip_runtime.h>

typedef float v2f __attribute__((ext_vector_type(2)));
typedef float v8f __attribute__((ext_vector_type(8)));

#define HID   64
#define ALPHA 0.2f

__device__ __forceinline__ float leaky(float x) { return x > 0.0f ? x : ALPHA * x; }

// ---------------- utility kernels ----------------
__global__ void k_fill(float* __restrict__ p, float v, int n) {
  int i = blockIdx.x * blockDim.x + threadIdx.x;
  if (i < n) p[i] = v;
}

__global__ void k_count(const int* __restrict__ dst, float* __restrict__ deg, int n) {
  int i = blockIdx.x * blockDim.x + threadIdx.x;
  if (i < n) atomicAdd(&deg[dst[i]], 1.0f);
}

__global__ void k_rsqrt(float* __restrict__ p, int n) {
  int i = blockIdx.x * blockDim.x + threadIdx.x;
  if (i < n) p[i] = rsqrtf(p[i]);
}

// ---------------- WMMA f32 GEMM: A[rows x K] @ W[K x 64] ----------------
// One wave computes one 16x16 output tile via V_WMMA_F32_16X16X4_F32.
// Half-select is folded into the load ADDRESS (b64 per lane) so the inner
// loop is pure load + wmma, no per-step cndmask.
// MODE 0: out = A@W              (GCN projection, bias added later)
// MODE 1: out = leaky(A@W + b)   (input linear layers)
template <int MODE>
__global__ void k_gemm_wmma(const float* __restrict__ A, const float* __restrict__ W,
                            const float* __restrict__ bias, float* __restrict__ out,
                            int rows, int K) {
  const int lane   = threadIdx.x & 31;
  const int wv     = threadIdx.x >> 5;
  const int tile   = blockIdx.x * (blockDim.x >> 5) + wv;
  const int nTiles = (rows >> 4) * (HID / 16);
  if (tile >= nTiles) return;                 // wave-uniform: EXEC stays all-1s

  const int bm   = tile >> 2;                 // 4 column tiles (HID=64)
  const int bn   = tile & 3;
  const int r0   = bm << 4;
  const int c0   = bn << 4;
  const int half = lane >> 4;                 // 0: K pair {k,k+1}; 1: {k+2,k+3}
  const int lm   = lane & 15;
  const int col  = c0 + lm;

  // ISA 16x4 f32 A layout: lane lm holds row r0+lm; lanes 16-31 take K+2..K+3.
  const float* __restrict__ apair = A + (long long)(r0 + lm) * K + 2 * half;
  // ISA 4x16 f32 B layout: lane holds column col; K pair selected by half.
  const float* __restrict__ bbase = W + (long long)(2 * half) * HID + col;

  v8f acc = {};
#pragma unroll 8
  for (int k = 0; k < K; k += 4) {
    const v2f a = *(const v2f*)(apair + k);           // {A[row][kk], A[row][kk+1]}
    v2f b;
    b.x = bbase[(long long)k * HID];                  // W[kk  ][col]
    b.y = bbase[(long long)k * HID + HID];            // W[kk+1][col]
    acc = __builtin_amdgcn_wmma_f32_16x16x4_f32(
        /*neg_a=*/false, a, /*neg_b=*/false, b,
        /*c_mod=*/(short)0, acc, /*reuse_a=*/false, /*reuse_b=*/false);
  }

  const float bv = (MODE == 1) ? bias[col] : 0.0f;
#pragma unroll
  for (int v = 0; v < 8; ++v) {
    const int row = r0 + v + half * 8;                // C/D VGPR layout
    float val = acc[v];
    if (MODE == 1) val = leaky(val + bv);
    out[(long long)row * HID + col] = val;
  }
}

// ---------------- edge scatter: agg[d] += t[s] * dinv[s]*dinv[d] ----------------
// 16 threads per edge, float4 per thread: coalesced gather + f32 global atomics.
__global__ void k_edge_scatter(const int* __restrict__ src, const int* __restrict__ dst,
                               const float* __restrict__ dinv, const float* __restrict__ t,
                               float* __restrict__ agg, int nEdge) {
  const long long tid = (long long)blockIdx.x * blockDim.x + threadIdx.x;
  if (tid >= (long long)nEdge * 16) return;
  const int e = (int)(tid >> 4);
  const int f = ((int)tid & 15) * 4;
  const int s = src[e];
  const int d = dst[e];
  const float w = dinv[s] * dinv[d];
  const float4 v = *(const float4*)(t + (long long)s * HID + f);
  float* o = agg + (long long)d * HID + f;
  atomicAdd(o + 0, v.x * w);
  atomicAdd(o + 1, v.y * w);
  atomicAdd(o + 2, v.z * w);
  atomicAdd(o + 3, v.w * w);
}

// ---------------- fused self-loop + bias + LeakyReLU ----------------
__global__ void k_finalize(const float* __restrict__ agg, const float* __restrict__ t,
                           const float* __restrict__ dinv, const float* __restrict__ bias,
                           float* __restrict__ out, int n) {
  const long long tid = (long long)blockIdx.x * blockDim.x + threadIdx.x;
  if (tid >= (long long)n * 16) return;
  const int i = (int)(tid >> 4);
  const int f = ((int)tid & 15) * 4;
  const float di = dinv[i];
  const float w = di * di;
  const float4 a  = *(const float4*)(agg + (long long)i * HID + f);
  const float4 tv = *(const float4*)(t + (long long)i * HID + f);
  float4 r;
  r.x = leaky(a.x + tv.x * w + bias[f + 0]);
  r.y = leaky(a.y + tv.y * w + bias[f + 1]);
  r.z = leaky(a.z + tv.z * w + bias[f + 2]);
  r.w = leaky(a.w + tv.w * w + bias[f + 3]);
  *(float4*)(out + (long long)i * HID + f) = r;
}

// ---------------- classifier (64 -> 2) + log_softmax ----------------
__global__ void k_classify(const float* __restrict__ z, const float* __restrict__ W,
                           const float* __restrict__ b, float* __restrict__ out, int n) {
  const int i = blockIdx.x * blockDim.x + threadIdx.x;
  if (i >= n) return;
  const float* zi = z + (long long)i * HID;
  float l0 = b[0], l1 = b[1];
#pragma unroll 8
  for (int k = 0; k < HID; ++k) {
    const float zv = zi[k];
    l0 += zv * W[k * 2 + 0];
    l1 += zv * W[k * 2 + 1];
  }
  const float m = fmaxf(l0, l1);
  const float lse = m + logf(expf(l0 - m) + expf(l1 - m));
  out[2 * i + 0] = l0 - lse;
  out[2 * i + 1] = l1 - lse;
}

// ---------------- orchestration ----------------
static inline int cdiv(long long a, int b) { return (int)((a + b - 1) / b); }

extern "C" void kernel_launch(void* const* d_in, const int* in_sizes, int n_in,
                              void* d_out, int out_size, void* d_ws, size_t ws_size,
                              hipStream_t stream) {
  const float* x      = (const float*)d_in[0];
  const float* mx     = (const float*)d_in[1];
  const int*   ei     = (const int*)d_in[2];
  const int*   mei    = (const int*)d_in[3];
  const float* W_lin  = (const float*)d_in[4];
  const float* b_lin  = (const float*)d_in[5];
  const float* W_mlin = (const float*)d_in[6];
  const float* b_mlin = (const float*)d_in[7];
  const float* Ws     = (const float*)d_in[8];
  const float* bs     = (const float*)d_in[9];
  const float* W_mg   = (const float*)d_in[10];
  const float* b_mg   = (const float*)d_in[11];
  const float* W_cls  = (const float*)d_in[12];
  const float* b_cls  = (const float*)d_in[13];

  const int F_IN = 128;
  const int N  = in_sizes[0] / F_IN;   // 100000
  const int M  = in_sizes[1] / F_IN;   // 100000
  const int E  = in_sizes[2] / 2;      // 1600000
  const int ME = in_sizes[3] / 2;      // 300000 (explicit meta edges incl. arange(N+M))
  const int NM = N + M;

  const int* src  = ei;
  const int* dst  = ei + E;
  const int* msrc = mei;
  const int* mdst = mei + ME;

  char* ws = (char*)d_ws;
  float* buf_h   = (float*)ws; ws += (size_t)NM * HID * sizeof(float); // h rows [0,N), mh rows [N,NM)
  float* buf_t   = (float*)ws; ws += (size_t)NM * HID * sizeof(float); // GEMM output scratch
  float* buf_agg = (float*)ws; ws += (size_t)NM * HID * sizeof(float); // scatter accumulator
  float* dinv_g  = (float*)ws; ws += (size_t)N * sizeof(float);
  float* dinv_m  = (float*)ws;

  // --- symmetric normalization coefficients (self-loops => deg starts at 1) ---
  k_fill <<<cdiv(N, 256), 256, 0, stream>>>(dinv_g, 1.0f, N);
  k_count<<<cdiv(E, 256), 256, 0, stream>>>(dst, dinv_g, E);
  k_rsqrt<<<cdiv(N, 256), 256, 0, stream>>>(dinv_g, N);
  k_fill <<<cdiv(NM, 256), 256, 0, stream>>>(dinv_m, 1.0f, NM);
  k_count<<<cdiv(ME, 256), 256, 0, stream>>>(mdst, dinv_m, ME);
  k_rsqrt<<<cdiv(NM, 256), 256, 0, stream>>>(dinv_m, NM);

  // --- input linear layers (f32 WMMA) ---
  const int tilesN  = (N >> 4) * 4;
  const int tilesM  = (M >> 4) * 4;
  const int tilesNM = (NM >> 4) * 4;
  k_gemm_wmma<1><<<cdiv(tilesN, 8), 256, 0, stream>>>(x,  W_lin,  b_lin,  buf_h, N, F_IN);
  k_gemm_wmma<1><<<cdiv(tilesM, 8), 256, 0, stream>>>(mx, W_mlin, b_mlin, buf_h + (size_t)N * HID, M, F_IN);

  // --- 3 GCN layers on the graph part ---
  for (int l = 0; l < 3; ++l) {
    k_gemm_wmma<0><<<cdiv(tilesN, 8), 256, 0, stream>>>(buf_h, Ws + (size_t)l * HID * HID,
                                                        nullptr, buf_t, N, HID);
    k_fill<<<cdiv((long long)N * HID, 256), 256, 0, stream>>>(buf_agg, 0.0f, N * HID);
    k_edge_scatter<<<cdiv((long long)E * 16, 256), 256, 0, stream>>>(src, dst, dinv_g, buf_t, buf_agg, E);
    k_finalize<<<cdiv((long long)N * 16, 256), 256, 0, stream>>>(buf_agg, buf_t, dinv_g,
                                                                 bs + (size_t)l * HID, buf_h, N);
  }

  // --- meta GCN over concat(h, mh) = buf_h[0, NM) ---
  k_gemm_wmma<0><<<cdiv(tilesNM, 8), 256, 0, stream>>>(buf_h, W_mg, nullptr, buf_t, NM, HID);
  k_fill<<<cdiv((long long)NM * HID, 256), 256, 0, stream>>>(buf_agg, 0.0f, NM * HID);
  k_edge_scatter<<<cdiv((long long)ME * 16, 256), 256, 0, stream>>>(msrc, mdst, dinv_m, buf_t, buf_agg, ME);
  k_finalize<<<cdiv((long long)NM * 16, 256), 256, 0, stream>>>(buf_agg, buf_t, dinv_m, b_mg, buf_h, NM);

  // --- classifier + log_softmax ---
  k_classify<<<cdiv(NM, 256), 256, 0, stream>>>(buf_h, W_cls, b_cls, (float*)d_out, NM);
}